// GCNLayer_13271448944838
// MI455X (gfx1250) — compile-verified
//
#include <hip/hip_runtime.h>

typedef float v2f __attribute__((ext_vector_type(2)));
typedef float v8f __attribute__((ext_vector_type(8)));

#define N_NODES 100000
#define N_EDGES 3200000
#define D       128
#define EPS     1e-5f

// ---- workspace layout (in floats) ----
// [0, 12.8M)  agg        (node-sum accumulator, later divided -> mean in place)
// then        deg        (N_NODES)
// then        colsum     (D)   per-feature sum of h = mean@W + b
// then        colsq      (D)   per-feature sum of h^2
// then        scale      (D)   gamma * rsqrt(var+eps)
// then        shift      (D)   beta - mu*scale
static constexpr size_t AGG_F      = (size_t)N_NODES * D;      // 12,800,000
static constexpr size_t DEG_OFF    = AGG_F;                    // 12,800,000
static constexpr size_t COLSUM_OFF = DEG_OFF + N_NODES;        // 12,900,000
static constexpr size_t COLSQ_OFF  = COLSUM_OFF + D;           // 12,900,128
static constexpr size_t SCALE_OFF  = COLSQ_OFF + D;            // 12,900,256
static constexpr size_t SHIFT_OFF  = SCALE_OFF + D;            // 12,900,384
static constexpr size_t ZERO_F     = SCALE_OFF;                // zero agg+deg+stats

// -------------------- kernel 0: zero accumulators --------------------
__global__ void k_zero(float4* __restrict__ ws4, long n4) {
    long i = (long)blockIdx.x * blockDim.x + threadIdx.x;
    if (i < n4) ws4[i] = make_float4(0.f, 0.f, 0.f, 0.f);
}

// -------------------- kernel 1: edge scatter (segment_sum) --------------------
// one wave per edge: 32 lanes x float4 = 128 floats; hardware f32 atomics.
__global__ void __launch_bounds__(256) k_edge_agg(
        const float* __restrict__ feat, const int* __restrict__ src,
        const int* __restrict__ dst, float* __restrict__ agg,
        float* __restrict__ deg) {
    long wave = ((long)blockIdx.x * blockDim.x + threadIdx.x) >> 5;
    int  lane = threadIdx.x & 31;
    if (wave >= N_EDGES) return;
    int s = src[wave];
    int d = dst[wave];
    float4 v = ((const float4*)(feat + (size_t)s * D))[lane];
    float* arow = agg + (size_t)d * D + lane * 4;
    unsafeAtomicAdd(arow + 0, v.x);
    unsafeAtomicAdd(arow + 1, v.y);
    unsafeAtomicAdd(arow + 2, v.z);
    unsafeAtomicAdd(arow + 3, v.w);
    if (lane == 0) unsafeAtomicAdd(deg + d, 1.0f);
}

// -------------------- kernel 2: agg /= max(deg,1) in place --------------------
__global__ void k_mean(float4* __restrict__ agg4, const float* __restrict__ deg) {
    long i = (long)blockIdx.x * blockDim.x + threadIdx.x;   // one float4 each
    if (i >= (long)N_NODES * (D / 4)) return;
    int row = (int)(i >> 5);                                // 32 float4 per row
    float inv = 1.0f / fmaxf(deg[row], 1.0f);
    float4 v = agg4[i];
    v.x *= inv; v.y *= inv; v.z *= inv; v.w *= inv;
    agg4[i] = v;
}

// -------------------- kernel 3: WMMA GEMM  h = mean @ W + b --------------------
// block = 256 threads = 8 waves; block handles 16 rows; wave w -> cols [16w,16w+16)
// K looped in steps of 4 with V_WMMA_F32_16X16X4_F32 (32 wmma per wave).
__global__ void __launch_bounds__(256) k_gemm_wmma(
        const float* __restrict__ h, const float* __restrict__ W,
        const float* __restrict__ bias, float* __restrict__ out,
        float* __restrict__ colsum, float* __restrict__ colsq) {
    const int lane   = threadIdx.x & 31;
    const int waveId = threadIdx.x >> 5;          // 0..7 -> N tile
    const int m_base = blockIdx.x * 16;           // 6250 blocks * 16 = 100000 exact
    const int n_base = waveId * 16;
    const int half   = lane >> 4;                 // 0: K={0,1}, 1: K={2,3}
    const int l      = lane & 15;

    // A: lane l holds row (m_base+l), two consecutive K at offset half*2
    const float* arow = h + (size_t)(m_base + l) * D + half * 2;
    // B: lane l holds col (n_base+l), rows k0+half*2 and k0+half*2+1
    const float* bcol = W + (size_t)(half * 2) * D + (n_base + l);

    v8f c = {0.f, 0.f, 0.f, 0.f, 0.f, 0.f, 0.f, 0.f};
#pragma unroll
    for (int k0 = 0; k0 < D; k0 += 4) {
        v2f a, b;
        a.x = arow[k0];
        a.y = arow[k0 + 1];
        b.x = bcol[(size_t)k0 * D];
        b.y = bcol[(size_t)k0 * D + D];
        c = __builtin_amdgcn_wmma_f32_16x16x4_f32(
                /*neg_a=*/false, a, /*neg_b=*/false, b,
                /*c_mod=*/(short)0, c, /*reuse_a=*/false, /*reuse_b=*/false);
    }

    // epilogue: + bias, store h_lin into d_out, accumulate BN stats
    const int   n  = n_base + l;
    const float bn = bias[n];
    float s = 0.f, s2 = 0.f;
#pragma unroll
    for (int r = 0; r < 8; ++r) {
        int   m   = m_base + half * 8 + r;        // VGPR r: M=r (lanes<16), M=r+8
        float val = c[r] + bn;
        out[(size_t)m * D + n] = val;
        s  += val;
        s2 += val * val;
    }
    unsafeAtomicAdd(&colsum[n], s);
    unsafeAtomicAdd(&colsq[n], s2);
}

// -------------------- kernel 4: fold BN stats into scale/shift --------------------
__global__ void k_stats(const float* __restrict__ colsum, const float* __restrict__ colsq,
                        const float* __restrict__ gamma, const float* __restrict__ beta,
                        float* __restrict__ scale, float* __restrict__ shift) {
    int d = threadIdx.x;
    if (d >= D) return;
    const float invN = 1.0f / (float)N_NODES;
    float mu  = colsum[d] * invN;
    float var = colsq[d] * invN - mu * mu;        // biased variance
    float sc  = gamma[d] * rsqrtf(var + EPS);
    scale[d] = sc;
    shift[d] = beta[d] - mu * sc;
}

// -------------------- kernel 5: out = feat + relu(h*scale + shift) --------------------
__global__ void k_finalize(const float4* __restrict__ feat4, float4* __restrict__ out4,
                           const float* __restrict__ scale, const float* __restrict__ shift) {
    long i = (long)blockIdx.x * blockDim.x + threadIdx.x;   // one float4 each
    if (i >= (long)N_NODES * (D / 4)) return;
    int d4 = ((int)i & 31) * 4;
    float4 hv = out4[i];
    float4 fv = feat4[i];
    float4 o;
    o.x = fv.x + fmaxf(hv.x * scale[d4 + 0] + shift[d4 + 0], 0.f);
    o.y = fv.y + fmaxf(hv.y * scale[d4 + 1] + shift[d4 + 1], 0.f);
    o.z = fv.z + fmaxf(hv.z * scale[d4 + 2] + shift[d4 + 2], 0.f);
    o.w = fv.w + fmaxf(hv.w * scale[d4 + 3] + shift[d4 + 3], 0.f);
    out4[i] = o;
}

extern "C" void kernel_launch(void* const* d_in, const int* in_sizes, int n_in,
                              void* d_out, int out_size, void* d_ws, size_t ws_size,
                              hipStream_t stream) {
    const float* feature = (const float*)d_in[0];
    const int*   src     = (const int*)d_in[1];
    const int*   dst     = (const int*)d_in[2];
    const float* W       = (const float*)d_in[3];
    const float* b       = (const float*)d_in[4];
    const float* gamma   = (const float*)d_in[5];
    const float* beta    = (const float*)d_in[6];
    float*       out     = (float*)d_out;

    float* ws     = (float*)d_ws;
    float* agg    = ws;
    float* deg    = ws + DEG_OFF;
    float* colsum = ws + COLSUM_OFF;
    float* colsq  = ws + COLSQ_OFF;
    float* scale  = ws + SCALE_OFF;
    float* shift  = ws + SHIFT_OFF;

    // 0) zero agg + deg + colsum + colsq (ZERO_F is divisible by 4)
    {
        long n4 = (long)(ZERO_F / 4);
        int  blocks = (int)((n4 + 255) / 256);
        k_zero<<<blocks, 256, 0, stream>>>((float4*)ws, n4);
    }
    // 1) edge aggregation: one wave per edge
    {
        long waves  = (long)N_EDGES;
        int  blocks = (int)((waves * 32 + 255) / 256);     // 400,000
        k_edge_agg<<<blocks, 256, 0, stream>>>(feature, src, dst, agg, deg);
    }
    // 2) mean = agg / max(deg,1), in place
    {
        long n4 = (long)N_NODES * (D / 4);                 // 3,200,000
        int  blocks = (int)((n4 + 255) / 256);             // 12,500
        k_mean<<<blocks, 256, 0, stream>>>((float4*)agg, deg);
    }
    // 3) WMMA GEMM + bias + BN-stat accumulation; h_lin written to d_out
    {
        int blocks = N_NODES / 16;                         // 6,250 exact
        k_gemm_wmma<<<blocks, 256, 0, stream>>>(agg, W, b, out, colsum, colsq);
    }
    // 4) fold stats
    k_stats<<<1, 128, 0, stream>>>(colsum, colsq, gamma, beta, scale, shift);
    // 5) normalize + relu + residual (in place on d_out)
    {
        long n4 = (long)N_NODES * (D / 4);
        int  blocks = (int)((n4 + 255) / 256);
        k_finalize<<<blocks, 256, 0, stream>>>((const float4*)feature, (float4*)out,
                                               scale, shift);
    }
}